// SlotAttention_76519137345594
// MI455X (gfx1250) — compile-verified
//
#include <hip/hip_runtime.h>
#include <math.h>

#define B_ 64
#define N_ 4096
#define D_ 64
#define K_ 8
#define ITERS_ 3
#define SCALE_ 0.125f
#define LNEPS 1e-5f
#define PIT 68   // LDS row pitch (floats): 272B rows -> 16B aligned, bank-conflict free

typedef __attribute__((ext_vector_type(2))) float v2f;
typedef __attribute__((ext_vector_type(8))) float v8f;

// D(16x16,f32) = A(16x4,f32) * B(4x16,f32) + C
__device__ __forceinline__ v8f wmma_f32_4(float ax, float ay, float bx, float by, v8f c) {
  v2f a; a.x = ax; a.y = ay;
  v2f b; b.x = bx; b.y = by;
  return __builtin_amdgcn_wmma_f32_16x16x4_f32(false, a, false, b, (short)0, c, false, false);
}

#if defined(__AMDGCN__) && __has_builtin(__builtin_amdgcn_global_load_async_to_lds_b128)
#define ASYNC_LDS 1
typedef int i4 __attribute__((ext_vector_type(4)));
typedef __attribute__((address_space(1))) i4* gas_p;   // global int4*
typedef __attribute__((address_space(3))) i4* las_p;   // LDS int4*
__device__ __forceinline__ void async_cp16(const float* g, float* l) {
  // 16B global -> LDS, tracked by ASYNCcnt
  __builtin_amdgcn_global_load_async_to_lds_b128((gas_p)g, (las_p)l, 0, 0);
}
__device__ __forceinline__ void wait_async() {
#if __has_builtin(__builtin_amdgcn_s_wait_asynccnt)
  __builtin_amdgcn_s_wait_asynccnt(0);
#else
  asm volatile("s_wait_asynccnt 0x0" ::: "memory");
#endif
}
#else
#define ASYNC_LDS 0
__device__ __forceinline__ void wait_async() {}
#endif

// ---------------------------------------------------------------------------
// Kernel 1: xn = LayerNorm(x); k = xn@Wk^T; v = xn@Wv^T   (fp32 WMMA)
// grid = B * (N/64), block = 256 (8 waves). Waves 0-3 -> k rows, 4-7 -> v rows.
// ---------------------------------------------------------------------------
__global__ __launch_bounds__(256)
void ln_kv_kernel(const float* __restrict__ x, const float* __restrict__ Wk,
                  const float* __restrict__ Wv, const float* __restrict__ g_x,
                  const float* __restrict__ bt_x, float* __restrict__ kbuf,
                  float* __restrict__ vbuf) {
  __shared__ float xs[64 * PIT];
  __shared__ float wkl[64 * PIT];
  __shared__ float wvl[64 * PIT];
  const int tid = threadIdx.x;
  const int lane = tid & 31;
  const int wave = tid >> 5;
  const int blk = blockIdx.x;
  const int b = blk >> 6;           // 64 n-tiles per batch
  const int n0 = (blk & 63) * 64;

  const size_t xbase = ((size_t)b * N_ + n0) * D_;
#if ASYNC_LDS
#pragma unroll
  for (int i = 0; i < 4; ++i) {     // 1024 b128 transfers per 64x64 tile
    int t = tid + i * 256;
    int r = t >> 4, q = t & 15;
    async_cp16(x + xbase + r * 64 + q * 4, xs + r * PIT + q * 4);
    async_cp16(Wk + r * 64 + q * 4, wkl + r * PIT + q * 4);
    async_cp16(Wv + r * 64 + q * 4, wvl + r * PIT + q * 4);
  }
  wait_async();
#else
#pragma unroll
  for (int i = 0; i < 16; ++i) {
    int idx = tid + i * 256;
    int r = idx >> 6, c = idx & 63;
    xs[r * PIT + c] = x[xbase + idx];
    wkl[r * PIT + c] = Wk[idx];
    wvl[r * PIT + c] = Wv[idx];
  }
#endif
  __syncthreads();

  if (tid < 64) {                   // LayerNorm one row per thread
    float m = 0.f;
#pragma unroll
    for (int j = 0; j < 64; ++j) m += xs[tid * PIT + j];
    m *= (1.0f / 64.0f);
    float v = 0.f;
#pragma unroll
    for (int j = 0; j < 64; ++j) { float d = xs[tid * PIT + j] - m; v += d * d; }
    float rs = rsqrtf(v * (1.0f / 64.0f) + LNEPS);
#pragma unroll
    for (int j = 0; j < 64; ++j)
      xs[tid * PIT + j] = (xs[tid * PIT + j] - m) * rs * g_x[j] + bt_x[j];
  }
  __syncthreads();

  const float* wl = (wave < 4) ? wkl : wvl;
  float* out = (wave < 4) ? kbuf : vbuf;
  const int r0 = (wave & 3) * 16;
  const int lrow = lane & 15;
  const int kadd = (lane < 16) ? 0 : 2;   // A/B K-pairing per ISA layout
  const int radd = (lane < 16) ? 0 : 8;   // C row offset per ISA layout

#pragma unroll
  for (int ct = 0; ct < 4; ++ct) {        // 16-wide dout tiles
    v8f acc = {};
#pragma unroll
    for (int kk = 0; kk < 64; kk += 4) {
      float ax = xs[(r0 + lrow) * PIT + kk + kadd];
      float ay = xs[(r0 + lrow) * PIT + kk + kadd + 1];
      float bx = wl[(ct * 16 + lrow) * PIT + kk + kadd];   // B[kd,dout]=W[dout,kd]
      float by = wl[(ct * 16 + lrow) * PIT + kk + kadd + 1];
      acc = wmma_f32_4(ax, ay, bx, by, acc);
    }
#pragma unroll
    for (int j = 0; j < 8; ++j) {
      int row = r0 + j + radd;
      out[((size_t)b * N_ + n0 + row) * D_ + ct * 16 + lrow] = acc[j];
    }
  }
}

// ---------------------------------------------------------------------------
// Kernel 2: full slot-attention loop, one workgroup per batch. k/v from L2.
// Async double-buffered pipeline: v(s) load overlaps logits WMMAs,
// k(s+1) load overlaps P@V WMMAs.
// ---------------------------------------------------------------------------
__global__ __launch_bounds__(256)
void slot_attn_kernel(const float* __restrict__ noise,
                      const float* __restrict__ slots_mu,
                      const float* __restrict__ slots_ls,
                      const float* __restrict__ Wq,
                      const float* __restrict__ W_ih, const float* __restrict__ W_hh,
                      const float* __restrict__ b_ih, const float* __restrict__ b_hh,
                      const float* __restrict__ W1, const float* __restrict__ b1,
                      const float* __restrict__ W2, const float* __restrict__ b2,
                      const float* __restrict__ g_s, const float* __restrict__ bt_s,
                      const float* __restrict__ g_p, const float* __restrict__ bt_p,
                      const float* __restrict__ kbuf, const float* __restrict__ vbuf,
                      float* __restrict__ out) {
  __shared__ float bufK[128 * PIT];    // k tile -> flash acc_red -> GRU gates
  __shared__ float bufV[128 * PIT];    // v tile
  __shared__ float slots[K_ * D_];
  __shared__ float prev[K_ * D_];
  __shared__ float lnb[K_ * D_];
  __shared__ float updb[K_ * D_];
  __shared__ float hid[K_ * D_];
  __shared__ float qlds[16 * PIT];     // q padded to 16 rows (rows 8..15 zero)
  __shared__ float pst[8 * 16 * 17];   // per-wave P tile staging (C->A relayout)
  __shared__ float mred[8 * 16];
  __shared__ float lred[8 * 16];
  __shared__ float sumv[64];
  __shared__ float sumvp[4 * 64];

  const int tid = threadIdx.x;
  const int lane = tid & 31;
  const int w = tid >> 5;
  const int b = blockIdx.x;
  const int lrow = lane & 15;
  const int kadd = (lane < 16) ? 0 : 2;
  const int radd = (lane < 16) ? 0 : 8;
  const float inv_norm = 1.0f / (1.0f + (float)N_ * 1e-8f);
  const size_t bbase = (size_t)b * N_ * 64;
  const int steps = N_ / 128;

  for (int e = tid; e < 16 * PIT; e += 256) qlds[e] = 0.f;
  for (int e = tid; e < K_ * D_; e += 256) {
    int d = e & 63;
    slots[e] = slots_mu[d] + noise[(size_t)b * K_ * D_ + e] * expf(slots_ls[d]);
  }
  __syncthreads();

  float svreg = 0.f;  // partial column-sum of v (iter 0 only)

  for (int iter = 0; iter < ITERS_; ++iter) {
#if ASYNC_LDS
    // issue k(0) now: overlaps slots copy, LN and q projection
#pragma unroll
    for (int i = 0; i < 8; ++i) {
      int t = tid + i * 256;
      int r = t >> 4, q = t & 15;
      async_cp16(kbuf + bbase + r * 64 + q * 4, bufK + r * PIT + q * 4);
    }
#endif
    for (int e = tid; e < K_ * D_; e += 256) prev[e] = slots[e];
    __syncthreads();

    // lnb = LN(slots; g_s, bt_s)
    if (tid < K_) {
      float m = 0.f;
      for (int j = 0; j < 64; ++j) m += slots[tid * 64 + j];
      m *= (1.f / 64.f);
      float v = 0.f;
      for (int j = 0; j < 64; ++j) { float d = slots[tid * 64 + j] - m; v += d * d; }
      float rs = rsqrtf(v * (1.f / 64.f) + LNEPS);
      for (int j = 0; j < 64; ++j)
        lnb[tid * 64 + j] = (slots[tid * 64 + j] - m) * rs * g_s[j] + bt_s[j];
    }
    __syncthreads();

    // q = lnb @ Wq^T  (8x64, tiny -> VALU)
    for (int e = tid; e < K_ * D_; e += 256) {
      int row = e >> 6, d = e & 63;
      float acc = 0.f;
      for (int j = 0; j < 64; ++j) acc += lnb[row * 64 + j] * Wq[d * 64 + j];
      qlds[row * PIT + d] = acc;
    }
    __syncthreads();

    // ---- flash attention over N, each wave owns a 16-wide n stripe ----
    float mrun[8], lrun[8];
    v8f accv[4];
#pragma unroll
    for (int j = 0; j < 8; ++j) { mrun[j] = -__builtin_inff(); lrun[j] = 0.f; }
#pragma unroll
    for (int c = 0; c < 4; ++c) accv[c] = (v8f){};

    const int noff = w * 16;
    for (int s = 0; s < steps; ++s) {
      const size_t gT = bbase + (size_t)s * 128 * 64;
#if ASYNC_LDS
      wait_async();              // k(s) landed (issued last step / iter top)
#else
#pragma unroll
      for (int i = 0; i < 32; ++i) {
        int idx = tid + i * 256;
        bufK[(idx >> 6) * PIT + (idx & 63)] = kbuf[gT + idx];
      }
#endif
      __syncthreads();
#if ASYNC_LDS
#pragma unroll
      for (int i = 0; i < 8; ++i) {   // issue v(s): overlaps logits
        int t = tid + i * 256;
        int r = t >> 4, q = t & 15;
        async_cp16(vbuf + gT + r * 64 + q * 4, bufV + r * PIT + q * 4);
      }
#else
#pragma unroll
      for (int i = 0; i < 32; ++i) {
        int idx = tid + i * 256;
        bufV[(idx >> 6) * PIT + (idx & 63)] = vbuf[gT + idx];
      }
#endif

      // logits = q @ k_tile^T  (16 WMMAs, K=64)
      v8f cl = (v8f){};
#pragma unroll
      for (int kk = 0; kk < 64; kk += 4) {
        float ax = qlds[lrow * PIT + kk + kadd];
        float ay = qlds[lrow * PIT + kk + kadd + 1];
        float bx = bufK[(noff + lrow) * PIT + kk + kadd];     // B[kd,n]=k[n,kd]
        float by = bufK[(noff + lrow) * PIT + kk + kadd + 1];
        cl = wmma_f32_4(ax, ay, bx, by, cl);
      }
      // online softmax per row (half-wave shuffle reductions)
#pragma unroll
      for (int j = 0; j < 8; ++j) {
        float t = cl[j] * SCALE_;
        float rmax = t;
        rmax = fmaxf(rmax, __shfl_xor(rmax, 1, 32));
        rmax = fmaxf(rmax, __shfl_xor(rmax, 2, 32));
        rmax = fmaxf(rmax, __shfl_xor(rmax, 4, 32));
        rmax = fmaxf(rmax, __shfl_xor(rmax, 8, 32));
        float nm = fmaxf(mrun[j], rmax);
        float alpha = expf(mrun[j] - nm);
        float p = expf(t - nm);
        float rsum = p;
        rsum += __shfl_xor(rsum, 1, 32);
        rsum += __shfl_xor(rsum, 2, 32);
        rsum += __shfl_xor(rsum, 4, 32);
        rsum += __shfl_xor(rsum, 8, 32);
        lrun[j] = lrun[j] * alpha + rsum;
        mrun[j] = nm;
#pragma unroll
        for (int c = 0; c < 4; ++c) accv[c][j] *= alpha;
        pst[w * 272 + (j + radd) * 17 + lrow] = p;   // C-layout -> LDS
      }
#if ASYNC_LDS
      wait_async();              // v(s) landed
#endif
      __syncthreads();           // all waves done with bufK, v visible
#if ASYNC_LDS
      if (s + 1 < steps) {       // issue k(s+1): overlaps P@V
        const size_t gN = gT + 128 * 64;
#pragma unroll
        for (int i = 0; i < 8; ++i) {
          int t = tid + i * 256;
          int r = t >> 4, q = t & 15;
          async_cp16(kbuf + gN + r * 64 + q * 4, bufK + r * PIT + q * 4);
        }
      }
#endif
      if (iter == 0) {  // accumulate sum_n v[n][d] for the +1e-8 renorm term
        int d = tid & 63, q4 = tid >> 6;
#pragma unroll
        for (int r = 0; r < 32; ++r) svreg += bufV[(q4 * 32 + r) * PIT + d];
      }

      // acc += P @ V_tile  (A from pst, B rows = v rows)
#pragma unroll
      for (int c = 0; c < 4; ++c) {
        v8f a2 = accv[c];
#pragma unroll
        for (int kk = 0; kk < 16; kk += 4) {
          float ax = pst[w * 272 + lrow * 17 + kk + kadd];
          float ay = pst[w * 272 + lrow * 17 + kk + kadd + 1];
          float bx = bufV[(noff + kk + kadd) * PIT + c * 16 + lrow];
          float by = bufV[(noff + kk + kadd + 1) * PIT + c * 16 + lrow];
          a2 = wmma_f32_4(ax, ay, bx, by, a2);
        }
        accv[c] = a2;
      }
      __syncthreads();           // bufV free before next v issue
    }

    // ---- cross-wave flash combine (acc_red reuses bufK) ----
    if ((lane & 15) == 0) {
      int half = lane >> 4;
#pragma unroll
      for (int j = 0; j < 8; ++j) {
        mred[w * 16 + j + half * 8] = mrun[j];
        lred[w * 16 + j + half * 8] = lrun[j];
      }
    }
#pragma unroll
    for (int c = 0; c < 4; ++c)
#pragma unroll
      for (int j = 0; j < 8; ++j)
        bufK[(w * 16 + j + radd) * PIT + c * 16 + lrow] = accv[c][j];
    if (iter == 0) sumvp[(tid >> 6) * 64 + (tid & 63)] = svreg;
    __syncthreads();
    if (iter == 0 && tid < 64)
      sumv[tid] = sumvp[tid] + sumvp[64 + tid] + sumvp[128 + tid] + sumvp[192 + tid];
    __syncthreads();

    for (int e = tid; e < K_ * D_; e += 256) {
      int row = e >> 6, d = e & 63;
      float M = -__builtin_inff();
#pragma unroll
      for (int ww = 0; ww < 8; ++ww) M = fmaxf(M, mred[ww * 16 + row]);
      float den = 0.f, S = 0.f;
#pragma unroll
      for (int ww = 0; ww < 8; ++ww) {
        float ew = expf(mred[ww * 16 + row] - M);
        den += ew * lred[ww * 16 + row];
        S += ew * bufK[(ww * 16 + row) * PIT + d];
      }
      updb[e] = (S / den + 1e-8f * sumv[d]) * inv_norm;
    }
    __syncthreads();

    // ---- GRU (gate order r,z,n), weights streamed from L2; gates reuse bufK ----
    for (int o = tid; o < K_ * 192; o += 256) {
      int row = o / 192, c = o % 192;
      float ax = b_ih[c], ah = b_hh[c];
      for (int j = 0; j < 64; ++j) {
        ax += updb[row * 64 + j] * W_ih[c * 64 + j];
        ah += prev[row * 64 + j] * W_hh[c * 64 + j];
      }
      bufK[o] = ax;            // gx
      bufK[1536 + o] = ah;     // gh
    }
    __syncthreads();
    for (int e = tid; e < K_ * D_; e += 256) {
      int row = e >> 6, d = e & 63;
      float rx = bufK[row * 192 + d],       rh = bufK[1536 + row * 192 + d];
      float zx = bufK[row * 192 + 64 + d],  zh = bufK[1536 + row * 192 + 64 + d];
      float nx = bufK[row * 192 + 128 + d], nh = bufK[1536 + row * 192 + 128 + d];
      float r = 1.f / (1.f + expf(-(rx + rh)));
      float z = 1.f / (1.f + expf(-(zx + zh)));
      float n = tanhf(nx + r * nh);
      slots[e] = (1.f - z) * n + z * prev[e];
    }
    __syncthreads();

    // ---- residual MLP ----
    if (tid < K_) {
      float m = 0.f;
      for (int j = 0; j < 64; ++j) m += slots[tid * 64 + j];
      m *= (1.f / 64.f);
      float v = 0.f;
      for (int j = 0; j < 64; ++j) { float dd = slots[tid * 64 + j] - m; v += dd * dd; }
      float rs = rsqrtf(v * (1.f / 64.f) + LNEPS);
      for (int j = 0; j < 64; ++j)
        lnb[tid * 64 + j] = (slots[tid * 64 + j] - m) * rs * g_p[j] + bt_p[j];
    }
    __syncthreads();
    for (int e = tid; e < K_ * D_; e += 256) {
      int row = e >> 6, d = e & 63;
      float acc = b1[d];
      for (int j = 0; j < 64; ++j) acc += lnb[row * 64 + j] * W1[d * 64 + j];
      hid[e] = fmaxf(acc, 0.f);
    }
    __syncthreads();
    for (int e = tid; e < K_ * D_; e += 256) {
      int row = e >> 6, d = e & 63;
      float acc = b2[d];
      for (int j = 0; j < 64; ++j) acc += hid[row * 64 + j] * W2[d * 64 + j];
      slots[e] += acc;
    }
    __syncthreads();
  }

  for (int e = tid; e < K_ * D_; e += 256)
    out[(size_t)b * K_ * D_ + e] = slots[e];
}

extern "C" void kernel_launch(void* const* d_in, const int* in_sizes, int n_in,
                              void* d_out, int out_size, void* d_ws, size_t ws_size,
                              hipStream_t stream) {
  (void)in_sizes; (void)n_in; (void)out_size; (void)ws_size;
  const float* x     = (const float*)d_in[0];
  const float* noise = (const float*)d_in[1];
  const float* mu    = (const float*)d_in[2];
  const float* ls    = (const float*)d_in[3];
  const float* Wq    = (const float*)d_in[4];
  const float* Wk    = (const float*)d_in[5];
  const float* Wv    = (const float*)d_in[6];
  const float* Wih   = (const float*)d_in[7];
  const float* Whh   = (const float*)d_in[8];
  const float* bih   = (const float*)d_in[9];
  const float* bhh   = (const float*)d_in[10];
  const float* W1    = (const float*)d_in[11];
  const float* b1    = (const float*)d_in[12];
  const float* W2    = (const float*)d_in[13];
  const float* b2    = (const float*)d_in[14];
  const float* gx    = (const float*)d_in[15];
  const float* btx   = (const float*)d_in[16];
  const float* gs    = (const float*)d_in[17];
  const float* bts   = (const float*)d_in[18];
  const float* gp    = (const float*)d_in[19];
  const float* btp   = (const float*)d_in[20];

  float* kbuf = (float*)d_ws;
  float* vbuf = kbuf + (size_t)B_ * N_ * D_;
  float* outp = (float*)d_out;

  ln_kv_kernel<<<dim3(B_ * (N_ / 64)), dim3(256), 0, stream>>>(x, Wk, Wv, gx, btx, kbuf, vbuf);
  slot_attn_kernel<<<dim3(B_), dim3(256), 0, stream>>>(noise, mu, ls, Wq, Wih, Whh, bih, bhh,
                                                       W1, b1, W2, b2, gs, bts, gp, btp,
                                                       kbuf, vbuf, outp);
}